// SGC_15556371546772
// MI455X (gfx1250) — compile-verified
//
#include <hip/hip_runtime.h>
#include <math.h>

typedef __attribute__((ext_vector_type(2))) float v2f;
typedef __attribute__((ext_vector_type(8))) float v8f;

#define FIN 256
#define FOUT 64

// ---------------------------------------------------------------------------
// 1) deg[n] = 1.0 (self-loop weight)
// ---------------------------------------------------------------------------
__global__ void sgc_init_deg(float* __restrict__ deg, int N) {
  int i = blockIdx.x * blockDim.x + threadIdx.x;
  if (i < N) deg[i] = 1.0f;
}

// ---------------------------------------------------------------------------
// 2) w[e] = mean(edge_attr[e,:8]); deg[dst[e]] += w[e]
// ---------------------------------------------------------------------------
__global__ void sgc_edge_weight_deg(const float* __restrict__ edge_attr,
                                    const int* __restrict__ dst,
                                    float* __restrict__ w,
                                    float* __restrict__ deg,
                                    int E) {
  int e = blockIdx.x * blockDim.x + threadIdx.x;
  if (e >= E) return;
  const float4* ea = (const float4*)(edge_attr + (size_t)e * 8);
  float4 a = ea[0];
  float4 b = ea[1];
  float s = (a.x + a.y + a.z + a.w + b.x + b.y + b.z + b.w) * 0.125f;
  w[e] = s;
  __hip_atomic_fetch_add(deg + dst[e], s, __ATOMIC_RELAXED, __HIP_MEMORY_SCOPE_AGENT);
}

// ---------------------------------------------------------------------------
// 3) deg -> dinv (in place): dinv = deg>0 ? rsqrt(deg) : 0
// ---------------------------------------------------------------------------
__global__ void sgc_dinv(float* __restrict__ deg, int N) {
  int i = blockIdx.x * blockDim.x + threadIdx.x;
  if (i < N) {
    float d = deg[i];
    deg[i] = (d > 0.0f) ? rsqrtf(d) : 0.0f;
  }
}

// ---------------------------------------------------------------------------
// 4a) self-loop term, also initializes h_out:  h_out[n,:] = dinv[n]^2 * h_in[n,:]
//     processed as float4 (64 float4 per row)
// ---------------------------------------------------------------------------
__global__ void sgc_self_loop(const float* __restrict__ h_in,
                              float* __restrict__ h_out,
                              const float* __restrict__ dinv,
                              int N) {
  size_t i = (size_t)blockIdx.x * blockDim.x + threadIdx.x;   // float4 index
  size_t total = (size_t)N * (FIN / 4);
  if (i >= total) return;
  int node = (int)(i >> 6);                                   // i / 64
  float di = dinv[node];
  float s = di * di;
  float4 v = ((const float4*)h_in)[i];
  float4 r;
  r.x = v.x * s; r.y = v.y * s; r.z = v.z * s; r.w = v.w * s;
  ((float4*)h_out)[i] = r;
}

// ---------------------------------------------------------------------------
// 4b) edge scatter:  h_out[dst,:] += dinv[src]*w[e]*dinv[dst] * h_in[src,:]
//     one wave32 per edge, 8 floats (2 x float4 loads) per lane
// ---------------------------------------------------------------------------
__global__ void sgc_edge_prop(const float* __restrict__ h_in,
                              float* __restrict__ h_out,
                              const int* __restrict__ src,
                              const int* __restrict__ dst,
                              const float* __restrict__ w,
                              const float* __restrict__ dinv,
                              int E) {
  int e = blockIdx.x * 8 + (threadIdx.x >> 5);   // 8 edges per 256-thread block
  if (e >= E) return;
  int lane = threadIdx.x & 31;
  int s = src[e];
  int d = dst[e];
  float nrm = dinv[s] * w[e] * dinv[d];
  const float4* pin = (const float4*)(h_in + (size_t)s * FIN) + lane * 2;
  float* pout = h_out + (size_t)d * FIN + lane * 8;
  float4 v0 = pin[0];
  float4 v1 = pin[1];
  __hip_atomic_fetch_add(pout + 0, nrm * v0.x, __ATOMIC_RELAXED, __HIP_MEMORY_SCOPE_AGENT);
  __hip_atomic_fetch_add(pout + 1, nrm * v0.y, __ATOMIC_RELAXED, __HIP_MEMORY_SCOPE_AGENT);
  __hip_atomic_fetch_add(pout + 2, nrm * v0.z, __ATOMIC_RELAXED, __HIP_MEMORY_SCOPE_AGENT);
  __hip_atomic_fetch_add(pout + 3, nrm * v0.w, __ATOMIC_RELAXED, __HIP_MEMORY_SCOPE_AGENT);
  __hip_atomic_fetch_add(pout + 4, nrm * v1.x, __ATOMIC_RELAXED, __HIP_MEMORY_SCOPE_AGENT);
  __hip_atomic_fetch_add(pout + 5, nrm * v1.y, __ATOMIC_RELAXED, __HIP_MEMORY_SCOPE_AGENT);
  __hip_atomic_fetch_add(pout + 6, nrm * v1.z, __ATOMIC_RELAXED, __HIP_MEMORY_SCOPE_AGENT);
  __hip_atomic_fetch_add(pout + 7, nrm * v1.w, __ATOMIC_RELAXED, __HIP_MEMORY_SCOPE_AGENT);
}

// ---------------------------------------------------------------------------
// 5) out = log_softmax(relu(h @ W + b))  via V_WMMA_F32_16X16X4_F32
//    block = 128 threads = 4 waves; each wave -> one 16x16 tile; 4 waves span
//    the 64 output columns; 16 rows per block. log-softmax via LDS staging.
//
//    WMMA f32 16x16x4 lane layout (wave32):
//      A (16x4):  lane l -> M = l&15, holds K = 2*(l>>4) .. +1  in 2 VGPRs
//      B (4x16):  lane l -> N = l&15, holds K = 2*(l>>4) .. +1  in 2 VGPRs
//      C/D:       lane l -> N = l&15, VGPR r -> M = r + 8*(l>>4)
// ---------------------------------------------------------------------------
__global__ void sgc_gemm_logsoftmax(const float* __restrict__ h,
                                    const float* __restrict__ Wm,
                                    const float* __restrict__ bias,
                                    float* __restrict__ out,
                                    int N) {
  __shared__ float tile[16][FOUT + 4];
  __shared__ float row_lse[16];

  int row0  = blockIdx.x * 16;
  int wave  = threadIdx.x >> 5;     // 0..3 -> col tile
  int lane  = threadIdx.x & 31;
  int m16   = lane & 15;
  int khalf = lane >> 4;            // 0 or 1
  int col0  = wave * 16;

  // clamp A-row for possible partial last block (loads only; stores guarded)
  int arow_idx = row0 + m16;
  if (arow_idx >= N) arow_idx = N - 1;
  const float* arow = h + (size_t)arow_idx * FIN;

  v8f acc = {0.0f, 0.0f, 0.0f, 0.0f, 0.0f, 0.0f, 0.0f, 0.0f};

  for (int k = 0; k < FIN; k += 4) {
    int ka = k + 2 * khalf;
    v2f a, b;
    a.x = arow[ka];
    a.y = arow[ka + 1];
    b.x = Wm[(size_t)ka * FOUT + col0 + m16];
    b.y = Wm[(size_t)(ka + 1) * FOUT + col0 + m16];
    acc = __builtin_amdgcn_wmma_f32_16x16x4_f32(
        /*neg_a=*/false, a, /*neg_b=*/false, b,
        /*c_mod=*/(short)0, acc, /*reuse_a=*/false, /*reuse_b=*/false);
  }

  // bias + relu -> LDS tile
  float bcol = bias[col0 + m16];
#pragma unroll
  for (int r = 0; r < 8; ++r) {
    int m = r + 8 * khalf;
    float v = acc[r] + bcol;
    v = v > 0.0f ? v : 0.0f;
    tile[m][col0 + m16] = v;
  }
  __syncthreads();

  // per-row log-sum-exp (16 rows; tiny serial reduction per thread)
  if (threadIdx.x < 16) {
    int m = threadIdx.x;
    float mx = -3.402823466e+38f;
    for (int c = 0; c < FOUT; ++c) mx = fmaxf(mx, tile[m][c]);
    float se = 0.0f;
    for (int c = 0; c < FOUT; ++c) se += expf(tile[m][c] - mx);
    row_lse[m] = mx + logf(se);
  }
  __syncthreads();

  // write out: 16*64 values / 128 threads = 8 each
  for (int idx = threadIdx.x; idx < 16 * FOUT; idx += blockDim.x) {
    int m = idx >> 6;
    int c = idx & 63;
    int row = row0 + m;
    if (row < N) out[(size_t)row * FOUT + c] = tile[m][c] - row_lse[m];
  }
}

// ---------------------------------------------------------------------------
// Host-side launcher
// ---------------------------------------------------------------------------
extern "C" void kernel_launch(void* const* d_in, const int* in_sizes, int n_in,
                              void* d_out, int out_size, void* d_ws, size_t ws_size,
                              hipStream_t stream) {
  const float* x         = (const float*)d_in[0];   // [N, 256]
  const float* edge_attr = (const float*)d_in[1];   // [E, 8]
  const float* Wm        = (const float*)d_in[2];   // [256, 64]
  const float* bias      = (const float*)d_in[3];   // [64]
  const int*   edge_idx  = (const int*)d_in[4];     // [2, E]

  const int N = in_sizes[0] / FIN;
  const int E = in_sizes[1] / 8;
  const int* src = edge_idx;
  const int* dst = edge_idx + E;

  float* out = (float*)d_out;

  // workspace layout
  char* p = (char*)d_ws;
  float* w    = (float*)p;  p += sizeof(float) * (size_t)E;        // edge weights
  float* dinv = (float*)p;  p += sizeof(float) * (size_t)N;        // deg -> dinv (in place)
  float* hA   = (float*)p;  p += sizeof(float) * (size_t)N * FIN;  // ping
  float* hB   = (float*)p;                                         // pong

  const int T = 256;

  // normalization coefficients
  sgc_init_deg<<<(N + T - 1) / T, T, 0, stream>>>(dinv, N);
  sgc_edge_weight_deg<<<(E + T - 1) / T, T, 0, stream>>>(edge_attr, dst, w, dinv, E);
  sgc_dinv<<<(N + T - 1) / T, T, 0, stream>>>(dinv, N);

  const size_t nvec4   = (size_t)N * (FIN / 4);
  const int    selfGrd = (int)((nvec4 + T - 1) / T);
  const int    edgeGrd = (E + 7) / 8;   // 8 edges per 256-thread block

  // hop 1: x -> hA
  sgc_self_loop<<<selfGrd, T, 0, stream>>>(x, hA, dinv, N);
  sgc_edge_prop<<<edgeGrd, T, 0, stream>>>(x, hA, src, dst, w, dinv, E);
  // hop 2: hA -> hB
  sgc_self_loop<<<selfGrd, T, 0, stream>>>(hA, hB, dinv, N);
  sgc_edge_prop<<<edgeGrd, T, 0, stream>>>(hA, hB, src, dst, w, dinv, E);
  // hop 3: hB -> hA
  sgc_self_loop<<<selfGrd, T, 0, stream>>>(hB, hA, dinv, N);
  sgc_edge_prop<<<edgeGrd, T, 0, stream>>>(hB, hA, src, dst, w, dinv, E);

  // GEMM + bias + relu + log_softmax (WMMA f32)
  sgc_gemm_logsoftmax<<<(N + 15) / 16, 128, 0, stream>>>(hA, Wm, bias, out, N);
}